// SocialInteraction2_16716012716116
// MI455X (gfx1250) — compile-verified
//
#include <hip/hip_runtime.h>

typedef __attribute__((ext_vector_type(2))) float v2f;
typedef __attribute__((ext_vector_type(8))) float v8f;

#define PN    1024   // P
#define MD    64     // M_DIM
#define RD    32     // R_DIM
#define TI    16     // i-rows per block (WMMA M)
#define SROW  1028   // padded LDS row stride (floats); 1028%64=4 -> conflict-free A fetch
#define NTHR  512    // 16 waves (wave32)

__global__ __launch_bounds__(NTHR)
void social_tensor_kernel(const float* __restrict__ hidden,
                          const float* __restrict__ corr,
                          const int*   __restrict__ nei,
                          const float* __restrict__ W_rela,
                          const float* __restrict__ b_rela,
                          const float* __restrict__ W_att,
                          const float* __restrict__ b_att,
                          float* __restrict__ out)
{
    __shared__ float s[TI * SROW];          // score tile -> softmax weights (66KB)
    __shared__ float hv_nei[PN];            // hidden[j]·w_nei
    __shared__ float hv_hi[TI];             // hidden[i0+ii]·w_hi
    __shared__ float wr0[RD], wr1[RD], brr[RD], war[RD];
    __shared__ float whi[MD], wnei[MD];
    __shared__ float partial[3 * 4 * 8 * 32];  // K-quarter partial accumulators

    const int tid  = threadIdx.x;
    const int lane = tid & 31;
    const int wave = tid >> 5;
    const int i0   = blockIdx.x * TI;

    // ---- stage small weights into LDS (broadcast reads later) ----
    if (tid < RD) {
        wr0[tid] = W_rela[tid];           // W_rela[0][r]
        wr1[tid] = W_rela[RD + tid];      // W_rela[1][r]
        brr[tid] = b_rela[tid];
        war[tid] = W_att[tid];            // w_r
    }
    if (tid >= 64 && tid < 64 + MD) {
        int m = tid - 64;
        whi[m]  = W_att[RD + m];          // w_hi
        wnei[m] = W_att[RD + MD + m];     // w_nei
    }
    __syncthreads();

    // ---- phase 0: rank-1 score terms ----
    for (int j = tid; j < PN; j += NTHR) {
        const float* h = hidden + j * MD;
        float acc = 0.f;
        #pragma unroll 8
        for (int m = 0; m < MD; ++m) acc = fmaf(h[m], wnei[m], acc);
        hv_nei[j] = acc;
    }
    if (tid < TI) {
        const float* h = hidden + (i0 + tid) * MD;
        float acc = 0.f;
        #pragma unroll 8
        for (int m = 0; m < MD; ++m) acc = fmaf(h[m], whi[m], acc);
        hv_hi[tid] = acc;
    }
    __syncthreads();

    const float batt = b_att[0];

    // ---- phase A: scores for 16 x 1024 tile (VALU-bound stage) ----
    for (int p = tid; p < TI * PN; p += NTHR) {
        const int ii = p >> 10;
        const int j  = p & (PN - 1);
        const long idx = (long)(i0 + ii) * PN + j;
        const float c0 = corr[idx * 2 + 0];
        const float c1 = corr[idx * 2 + 1];
        float acc = 0.f;
        #pragma unroll
        for (int r = 0; r < RD; ++r) {
            float t = fmaf(c1, wr1[r], brr[r]);
            t = fmaf(c0, wr0[r], t);
            t = fmaxf(t, 0.f);                 // relu
            acc = fmaf(t, war[r], acc);        // · w_r
        }
        float sc = acc + hv_hi[ii] + hv_nei[j] + batt;
        float pv = (nei[idx] > 0) ? sc : 0.f;  // where(mask, scores, 0)
        pv = (pv == 0.f) ? -1e-6f : pv;        // where(Pos==0, -1e-6, Pos)
        s[ii * SROW + j] = pv;
    }
    __syncthreads();

    // ---- phase B: softmax over j, one row per wave (wave32 shuffles) ----
    {
        const int ii = wave;                   // 16 waves, 16 rows
        float* row = s + ii * SROW;
        float mx = -3.402823466e+38f;
        for (int j = lane; j < PN; j += 32) mx = fmaxf(mx, row[j]);
        #pragma unroll
        for (int off = 16; off > 0; off >>= 1) mx = fmaxf(mx, __shfl_xor(mx, off, 32));
        float sum = 0.f;
        for (int j = lane; j < PN; j += 32) {
            float e = expf(row[j] - mx);
            row[j] = e;
            sum += e;
        }
        #pragma unroll
        for (int off = 16; off > 0; off >>= 1) sum += __shfl_xor(sum, off, 32);
        const float inv = 1.f / sum;
        const int* nr = nei + (long)(i0 + ii) * PN;
        for (int j = lane; j < PN; j += 32) {
            float mf = (nr[j] > 0) ? 1.f : 0.f; // Pos * mask
            row[j] = row[j] * inv * mf;
        }
    }
    __syncthreads();

    // ---- phase C: out(16x64) = W(16x1024) @ hidden(1024x64), f32 WMMA ----
    // wave -> (N-tile = wave&3, K-quarter = wave>>2)
    const int nt   = wave & 3;
    const int n0   = nt * 16;
    const int kq   = wave >> 2;
    const int kbeg = kq * (PN / 4);
    const int kend = kbeg + (PN / 4);

    v8f acc = {};
    const int arow = (lane & 15) * SROW;       // A: M = lane&15
    const int koff = (lane < 16) ? 0 : 2;      // lanes 16-31 hold K+2/K+3
    const int ncol = n0 + (lane & 15);         // B/D: N = lane&15
    for (int k = kbeg; k < kend; k += 4) {
        v2f a, b;
        a.x = s[arow + k + koff + 0];
        a.y = s[arow + k + koff + 1];
        b.x = hidden[(k + koff + 0) * MD + ncol];
        b.y = hidden[(k + koff + 1) * MD + ncol];
        acc = __builtin_amdgcn_wmma_f32_16x16x4_f32(false, a, false, b,
                                                    (short)0, acc, false, false);
    }

    if (kq > 0) {
        #pragma unroll
        for (int g = 0; g < 8; ++g)
            partial[(((kq - 1) * 4 + nt) * 8 + g) * 32 + lane] = acc[g];
    }
    __syncthreads();
    if (kq == 0) {
        #pragma unroll
        for (int g = 0; g < 8; ++g) {
            float v = acc[g]
                    + partial[((0 * 4 + nt) * 8 + g) * 32 + lane]
                    + partial[((1 * 4 + nt) * 8 + g) * 32 + lane]
                    + partial[((2 * 4 + nt) * 8 + g) * 32 + lane];
            const int m = g + ((lane < 16) ? 0 : 8);   // D-layout: VGPR g -> M=g / g+8
            out[(long)(i0 + m) * MD + n0 + (lane & 15)] = v;
        }
    }
}

extern "C" void kernel_launch(void* const* d_in, const int* in_sizes, int n_in,
                              void* d_out, int out_size, void* d_ws, size_t ws_size,
                              hipStream_t stream) {
    (void)in_sizes; (void)n_in; (void)out_size; (void)d_ws; (void)ws_size;
    const float* hidden = (const float*)d_in[0];
    const float* corr   = (const float*)d_in[1];
    const int*   nei    = (const int*)d_in[2];
    const float* W_rela = (const float*)d_in[3];
    const float* b_rela = (const float*)d_in[4];
    const float* W_att  = (const float*)d_in[5];
    const float* b_att  = (const float*)d_in[6];
    float* outp = (float*)d_out;

    social_tensor_kernel<<<dim3(PN / TI), dim3(NTHR), 0, stream>>>(
        hidden, corr, nei, W_rela, b_rela, W_att, b_att, outp);
}